// TermReader_40261023432787
// MI455X (gfx1250) — compile-verified
//
#include <hip/hip_runtime.h>

// Problem: B=64, L=2048, H=256
//   q = states @ W1.T + b1; k = emb @ W2.T + b2; s = <k,q>; softmax; term.
// Rewrite: s[b,l] = emb[b,l] . (W2^T q_b) + (b2 . q_b)  -> no [B,L,H] GEMM,
// just two streaming passes over emb (128 MiB). Small GEMMs use WMMA f32.

#define Bsz 64
#define Lsz 2048
#define Hsz 256
#define NEG_INF (-1e30f)

typedef float v2f __attribute__((ext_vector_type(2)));
typedef float v8f __attribute__((ext_vector_type(8)));
typedef int   v4i_vs __attribute__((vector_size(16)));   // matches builtin param

#define AS1 __attribute__((address_space(1)))
#define AS3 __attribute__((address_space(3)))

#if defined(__AMDGCN__) && __has_builtin(__builtin_amdgcn_global_load_async_to_lds_b128)
#define HAVE_ASYNC_LDS 1
#if __has_builtin(__builtin_amdgcn_s_wait_asynccnt)
#define ASYNC_WAIT(n) __builtin_amdgcn_s_wait_asynccnt(n)
#else
#define ASYNC_WAIT(n) asm volatile("s_wait_asynccnt %0" ::"n"(n) : "memory")
#endif
#endif

// ---------------- WMMA f32 16x16x4 GEMM kernels (one wave per 16x16 tile) ---

// Q[64,256] = S[64,259] @ W1[256,259]^T + b1     (K = 259 = 64*4 + 3)
__global__ __launch_bounds__(128)
void qgemm_kernel(const float* __restrict__ S, const float* __restrict__ W1,
                  const float* __restrict__ b1, float* __restrict__ Q)
{
    const int LDK = 259;
    int wave = threadIdx.x >> 5;
    int lane = threadIdx.x & 31;
    int tile = blockIdx.x * 4 + wave;     // 64 tiles: 4 (M) x 16 (N)
    int tm   = (tile >> 4) * 16;
    int tn   = (tile & 15) * 16;
    int half = lane >> 4;                 // 0: K lanes {0,1}, 1: {2,3}
    int mr   = lane & 15;
    int koff = half * 2;

    const float* arow = S  + (size_t)(tm + mr) * LDK;  // A[m, :]
    const float* brow = W1 + (size_t)(tn + mr) * LDK;  // B[k,n] = W1[n,k]

    v8f acc = {};
#pragma unroll 4
    for (int k0 = 0; k0 < 256; k0 += 4) {   // branch-free main loop
        v2f a, b;
        a[0] = arow[k0 + koff];
        a[1] = arow[k0 + koff + 1];
        b[0] = brow[k0 + koff];
        b[1] = brow[k0 + koff + 1];
        acc = __builtin_amdgcn_wmma_f32_16x16x4_f32(
                  false, a, false, b, (short)0, acc, false, false);
    }
    {   // tail: k = 256..258 valid, 259 masked (high-half lanes, j=1)
        int   k    = 256 + koff;
        float msk  = (koff < 2) ? 1.f : 0.f;
        int   k1   = (koff < 2) ? k + 1 : k;   // clamp: never touch index 259
        v2f a, b;
        a[0] = arow[k];
        a[1] = arow[k1] * msk;
        b[0] = brow[k];
        b[1] = brow[k1] * msk;
        acc = __builtin_amdgcn_wmma_f32_16x16x4_f32(
                  false, a, false, b, (short)0, acc, false, false);
    }
#pragma unroll
    for (int r = 0; r < 8; ++r) {
        int m = tm + r + 8 * half;
        int n = tn + mr;
        Q[m * Hsz + n] = acc[r] + b1[n];
    }
}

// V[64,256] = Q[64,256] @ W2[256,256]   (v_b = W2^T q_b)
__global__ __launch_bounds__(128)
void vgemm_kernel(const float* __restrict__ Q, const float* __restrict__ W2,
                  float* __restrict__ V)
{
    int wave = threadIdx.x >> 5;
    int lane = threadIdx.x & 31;
    int tile = blockIdx.x * 4 + wave;
    int tm   = (tile >> 4) * 16;
    int tn   = (tile & 15) * 16;
    int half = lane >> 4;
    int mr   = lane & 15;
    int koff = half * 2;

    const v2f*   arow = (const v2f*)(Q + (size_t)(tm + mr) * Hsz); // 8B aligned
    const float* bcol = W2 + (tn + mr);

    v8f acc = {};
#pragma unroll 4
    for (int k0 = 0; k0 < 256; k0 += 4) {
        v2f a = arow[(k0 + koff) >> 1];
        v2f b;
        b[0] = bcol[(size_t)(k0 + koff) * Hsz];
        b[1] = bcol[(size_t)(k0 + koff + 1) * Hsz];
        acc = __builtin_amdgcn_wmma_f32_16x16x4_f32(
                  false, a, false, b, (short)0, acc, false, false);
    }
#pragma unroll
    for (int r = 0; r < 8; ++r)
        V[(tm + r + 8 * half) * Hsz + (tn + mr)] = acc[r];
}

// c_b = b2 . Q[b]
__global__ __launch_bounds__(256)
void cdot_kernel(const float* __restrict__ Q, const float* __restrict__ b2,
                 float* __restrict__ C)
{
    __shared__ float red[256];
    int b = blockIdx.x, t = threadIdx.x;
    red[t] = Q[b * Hsz + t] * b2[t];
    __syncthreads();
    for (int s = 128; s > 0; s >>= 1) {
        if (t < s) red[t] += red[t + s];
        __syncthreads();
    }
    if (t == 0) C[b] = red[0];
}

// ---------------- Pass 1: scores[b,l] = emb[b,l].v_b + c_b (masked) --------
__global__ __launch_bounds__(256)
void scores_kernel(const float4* __restrict__ emb4, const float4* __restrict__ V4,
                   const float* __restrict__ C, const int* __restrict__ lengths,
                   float* __restrict__ scores)
{
    int b    = blockIdx.y;
    int lane = threadIdx.x & 31;
    int wave = threadIdx.x >> 5;

    // each lane keeps its 8-float slice of v_b in registers
    float4 v0 = V4[b * 64 + lane * 2];
    float4 v1 = V4[b * 64 + lane * 2 + 1];
    float  c  = C[b];
    int    len = lengths[b];

    int lbase = blockIdx.x * 256 + wave;   // 8 waves/block, 8 blocks in x
    for (int i = 0; i < 32; ++i) {
        int l = lbase + i * 8;
        const float4* e = emb4 + ((size_t)(b * Lsz + l)) * 64 + lane * 2;
        __builtin_prefetch((const void*)(e + 8 * 64), 0, 0); // next row, this lane
        float4 x0 = e[0], x1 = e[1];
        float p = x0.x * v0.x + x0.y * v0.y + x0.z * v0.z + x0.w * v0.w
                + x1.x * v1.x + x1.y * v1.y + x1.z * v1.z + x1.w * v1.w;
        p += __shfl_xor(p, 16, 32);
        p += __shfl_xor(p, 8, 32);
        p += __shfl_xor(p, 4, 32);
        p += __shfl_xor(p, 2, 32);
        p += __shfl_xor(p, 1, 32);
        if (lane == 0)
            scores[b * Lsz + l] = (l < len) ? (p + c) : NEG_INF;
    }
}

// ---------------- softmax stats (max, 1/Z) per batch; also zero d_out ------
__global__ __launch_bounds__(256)
void softstats_kernel(const float* __restrict__ scores, float* __restrict__ MZ,
                      float* __restrict__ out)
{
    __shared__ float red[256];
    int b = blockIdx.x, t = threadIdx.x;
    float m = -3.402823466e38f;
    for (int l = t; l < Lsz; l += 256) m = fmaxf(m, scores[b * Lsz + l]);
    red[t] = m;
    __syncthreads();
    for (int s = 128; s > 0; s >>= 1) {
        if (t < s) red[t] = fmaxf(red[t], red[t + s]);
        __syncthreads();
    }
    m = red[0];
    __syncthreads();
    float z = 0.f;
    for (int l = t; l < Lsz; l += 256) z += expf(scores[b * Lsz + l] - m);
    red[t] = z;
    __syncthreads();
    for (int s = 128; s > 0; s >>= 1) {
        if (t < s) red[t] += red[t + s];
        __syncthreads();
    }
    if (t == 0) { MZ[2 * b] = m; MZ[2 * b + 1] = 1.f / red[0]; }
    out[b * Hsz + t] = 0.f;   // zero output for pass-2 atomics
}

// ---------------- Pass 2: term_b += sum_l w[b,l] * emb[b,l,:] --------------
// Each block: one batch b, a 256-row slice of L. Rows staged into LDS via
// GLOBAL_LOAD_ASYNC_TO_LDS_B128, double-buffered 16-row chunks, overlapped
// with FMA accumulation via s_wait_asynccnt.
__global__ __launch_bounds__(256)
void term_kernel(const float* __restrict__ emb, const float* __restrict__ scores,
                 const float* __restrict__ MZ, float* __restrict__ out)
{
    __shared__ float w[256];
    int b = blockIdx.y, t = threadIdx.x;
    int l0 = blockIdx.x * 256;
    float m = MZ[2 * b], invZ = MZ[2 * b + 1];
    w[t] = expf(scores[b * Lsz + l0 + t] - m) * invZ;  // exp(NEG_INF-m)==0

    const float* gbase = emb + ((size_t)b * Lsz + l0) * Hsz;
    float acc = 0.f;

#if defined(HAVE_ASYNC_LDS)
    __shared__ float buf[2][16][Hsz];        // 2 x 16KB double buffer
    int lrow = t >> 6;                       // 0..3
    int lcol = (t & 63) * 4;                 // float4 column

    auto issue = [&](int c) {                // stage 16 rows of chunk c
#pragma unroll
        for (int p = 0; p < 4; ++p) {
            int row = lrow + p * 4;
            __builtin_amdgcn_global_load_async_to_lds_b128(
                (AS1 v4i_vs*)(gbase + (size_t)(c * 16 + row) * Hsz + lcol),
                (AS3 v4i_vs*)(&buf[c & 1][row][lcol]), 0, 0);
        }
    };

    issue(0);
    for (int c = 0; c < 16; ++c) {
        if (c < 15) {
            issue(c + 1);       // 4 new in flight on top of chunk c's 4
            ASYNC_WAIT(4);      // chunk c complete for this wave
        } else {
            ASYNC_WAIT(0);      // last chunk fully landed
        }
        __syncthreads();        // all waves' chunk-c data visible (w too, c==0)
#pragma unroll
        for (int i = 0; i < 16; ++i)
            acc += w[c * 16 + i] * buf[c & 1][i][t];
        __syncthreads();        // buffer free before chunk c+2 is issued
    }
#else
    __syncthreads();
    const float* e = gbase + t;
#pragma unroll 4
    for (int i = 0; i < 256; ++i) {
        __builtin_prefetch((const void*)(e + (size_t)(i + 16) * Hsz), 0, 0);
        acc += w[i] * e[(size_t)i * Hsz];
    }
#endif
    atomicAdd(&out[b * Hsz + t], acc);
}

// ---------------------------------------------------------------------------
extern "C" void kernel_launch(void* const* d_in, const int* in_sizes, int n_in,
                              void* d_out, int out_size, void* d_ws, size_t ws_size,
                              hipStream_t stream) {
    const float* states  = (const float*)d_in[0];   // [64,259]
    const float* emb     = (const float*)d_in[1];   // [64,2048,256]
    const int*   lengths = (const int*)  d_in[2];   // [64]
    const float* W1      = (const float*)d_in[3];   // [256,259]
    const float* b1      = (const float*)d_in[4];   // [256]
    const float* W2      = (const float*)d_in[5];   // [256,256]
    const float* b2      = (const float*)d_in[6];   // [256]
    float* out = (float*)d_out;                     // [64,256]

    char* ws = (char*)d_ws;
    float* Q      = (float*)(ws);                   //  64 KB
    float* V      = (float*)(ws + 65536);           //  64 KB
    float* C      = (float*)(ws + 131072);          //  256 B (padded)
    float* MZ     = (float*)(ws + 135168);          //  512 B (padded)
    float* scores = (float*)(ws + 139264);          // 512 KB

    qgemm_kernel<<<16, 128, 0, stream>>>(states, W1, b1, Q);
    vgemm_kernel<<<16, 128, 0, stream>>>(Q, W2, V);
    cdot_kernel <<<Bsz, 256, 0, stream>>>(Q, b2, C);
    scores_kernel<<<dim3(8, Bsz), 256, 0, stream>>>(
        (const float4*)emb, (const float4*)V, C, lengths, scores);
    softstats_kernel<<<Bsz, 256, 0, stream>>>(scores, MZ, out);
    term_kernel<<<dim3(8, Bsz), 256, 0, stream>>>(emb, scores, MZ, out);
}